// FraudGNN_73495480369648
// MI455X (gfx1250) — compile-verified
//
#include <hip/hip_runtime.h>
#include <math.h>

typedef __attribute__((ext_vector_type(2))) float v2f;
typedef __attribute__((ext_vector_type(8))) float v8f;
typedef __attribute__((ext_vector_type(4))) unsigned int u32x4;
typedef __attribute__((ext_vector_type(4))) int i32x4;
typedef __attribute__((ext_vector_type(8))) int i32x8;

#define LDS_STRIDE 130  // 128 data + 2 pad DWORDs per row: 8B-aligned, bank-stride 2

// ---------------- zero scratch ----------------
__global__ void zero_kernel(float* __restrict__ p, long n) {
    long i = (long)blockIdx.x * blockDim.x + threadIdx.x;
    long stride = (long)gridDim.x * blockDim.x;
    for (; i < n; i += stride) p[i] = 0.0f;
}

// ---------------- fused GEMM: P = x@W1_l, R = x@W1_r  (WMMA f32 16x16x4) ----------------
// block = 128 threads (4 waves). Block handles 64 rows; wave w handles rows base+16w..+15.
__global__ void __launch_bounds__(128)
gemm1_kernel(const float* __restrict__ x, const float* __restrict__ W1l,
             const float* __restrict__ W1r, float* __restrict__ P,
             float* __restrict__ R, int n) {
    __shared__ __align__(16) float sx [64 * LDS_STRIDE];  // x tile [row][k], stride 130
    __shared__ __align__(16) float swl[64 * LDS_STRIDE];  // W1_l transposed [n][k]
    __shared__ __align__(16) float swr[64 * LDS_STRIDE];  // W1_r transposed [n][k]

    const int tid  = threadIdx.x;
    const int base = blockIdx.x * 64;

#if __has_builtin(__builtin_amdgcn_tensor_load_to_lds)
    // ---- x tile via Tensor Data Mover: 1D copy of up to 64*128 f32, with
    //      LDS padding (2 DWORDs every 128) to produce the stride-130 layout,
    //      and tensor_dim0 clamped so OOB rows are hardware zero-filled.
    if (tid < 32) {  // wave 0 issues the DMA
        unsigned lds_off = (unsigned)(uintptr_t)(void*)sx;   // low 32 bits = LDS offset
        unsigned long long ga = (unsigned long long)(uintptr_t)(x + (long)base * 128);
        int rem = n - base; if (rem > 64) rem = 64;
        unsigned dim0 = (unsigned)rem * 128u;                // valid elements (zero-fill past)
        // D# group0: count=1 | lds_addr | global_addr[31:0] | global_addr[56:32], type=2
        u32x4 g0 = { 1u, lds_off, (unsigned)ga,
                     (unsigned)((ga >> 32) & 0x01FFFFFFu) | 0x80000000u };
        // D# group1: data_size=2(4B), pad_enable, pad_interval=6(128 DW), pad_amount=1(2 DW)
        //   tensor_dim0=dim0, tensor_dim1=1, tile_dim0=8192, tile_dim1=1, dim0_stride=8192
        i32x8 g1 = { (int)(0x20000u | (1u << 20) | (6u << 22) | (1u << 25)),
                     (int)((dim0 & 0xFFFFu) << 16),
                     (int)((dim0 >> 16) | (1u << 16)),
                     (int)(8192u << 16),
                     1,
                     8192,
                     0, 0 };
        i32x4 z4 = {0, 0, 0, 0};
#if defined(__clang_major__) && (__clang_major__ >= 23)
        i32x8 z8 = {0, 0, 0, 0, 0, 0, 0, 0};
        __builtin_amdgcn_tensor_load_to_lds(g0, g1, z4, z4, z8, 0);
#else
        __builtin_amdgcn_tensor_load_to_lds(g0, g1, z4, z4, 0);
#endif
        __builtin_amdgcn_s_wait_tensorcnt(0);
    }
#else
    for (int i = tid; i < 64 * 128; i += 128) {
        int r = i >> 7, c = i & 127;
        int gr = base + r;
        sx[r * LDS_STRIDE + c] = (gr < n) ? x[(long)gr * 128 + c] : 0.0f;
    }
#endif
    // stage weights transposed: W[k][n] -> sw[n][k]  (coalesced global reads)
    for (int i = tid; i < 128 * 64; i += 128) {
        int k = i >> 6, nn = i & 63;
        swl[nn * LDS_STRIDE + k] = W1l[i];
        swr[nn * LDS_STRIDE + k] = W1r[i];
    }
    __syncthreads();

    const int lane = tid & 31;
    const int w    = tid >> 5;
    const int half = lane >> 4;   // 0: lanes 0-15 (K=k,k+1)  1: lanes 16-31 (K=k+2,k+3)
    const int lr   = lane & 15;
    const int ar   = w * 16 + lr; // A row within tile
    const int koff = half * 2;

    const v8f zero8 = {0.f,0.f,0.f,0.f,0.f,0.f,0.f,0.f};
    v8f accP[4] = {zero8, zero8, zero8, zero8};
    v8f accR[4] = {zero8, zero8, zero8, zero8};

    for (int k = 0; k < 128; k += 4) {
        v2f a = *(const v2f*)&sx[ar * LDS_STRIDE + k + koff];
#pragma unroll
        for (int t = 0; t < 4; ++t) {
            v2f bl = *(const v2f*)&swl[(t * 16 + lr) * LDS_STRIDE + k + koff];
            accP[t] = __builtin_amdgcn_wmma_f32_16x16x4_f32(
                false, a, false, bl, (short)0, accP[t], false, false);
            v2f br = *(const v2f*)&swr[(t * 16 + lr) * LDS_STRIDE + k + koff];
            accR[t] = __builtin_amdgcn_wmma_f32_16x16x4_f32(
                false, a, false, br, (short)0, accR[t], false, false);
        }
    }

    // C/D layout: VGPR v -> M=v (lanes 0-15) / M=v+8 (lanes 16-31), N = lane&15
    const long rowbase = (long)base + w * 16 + half * 8;
    if (base + 64 <= n) {  // full tile: branch-free coalesced stores
#pragma unroll
        for (int t = 0; t < 4; ++t) {
            const int col = t * 16 + lr;
#pragma unroll
            for (int v = 0; v < 8; ++v) {
                long grow = rowbase + v;
                P[grow * 64 + col] = accP[t][v];
                R[grow * 64 + col] = accR[t][v];
            }
        }
    } else {               // ragged last tile
#pragma unroll
        for (int t = 0; t < 4; ++t) {
            const int col = t * 16 + lr;
#pragma unroll
            for (int v = 0; v < 8; ++v) {
                long grow = rowbase + v;
                if (grow < n) {
                    P[grow * 64 + col] = accP[t][v];
                    R[grow * 64 + col] = accR[t][v];
                }
            }
        }
    }
}

// ---------------- layer-1 edge aggregation: aggP[dst] += P[src]; deg[dst] += 1 ----------------
// 1 wave per edge iteration; lane handles 2 of 64 columns (coalesced 256B gather/edge).
// Software-pipelined grid-stride loop with prefetch of the next edge's P row.
__global__ void __launch_bounds__(256)
scatter1_kernel(const long long* __restrict__ ei, const float* __restrict__ P,
                float* __restrict__ aggP, float* __restrict__ deg, long E) {
    const int lane = threadIdx.x & 31;
    const int c = lane * 2;
    const long nwaves = (long)gridDim.x * 8;
    long e = (long)blockIdx.x * 8 + (threadIdx.x >> 5);
    if (e >= E) return;
    int s = (int)ei[e];
    int d = (int)ei[E + e];
    for (;;) {
        long en = e + nwaves;
        bool more = en < E;
        int sn = 0, dn = 0;
        if (more) {
            sn = (int)ei[en];
            dn = (int)ei[E + en];
            __builtin_prefetch(&P[(long)sn * 64 + c], 0, 1);  // global_prefetch_b8
        }
        float2 v = *(const float2*)&P[(long)s * 64 + c];
        unsafeAtomicAdd(&aggP[(long)d * 64 + c],     v.x);
        unsafeAtomicAdd(&aggP[(long)d * 64 + c + 1], v.y);
        if (lane == 0) unsafeAtomicAdd(&deg[d], 1.0f);
        if (!more) break;
        s = sn; d = dn; e = en;
    }
}

// ---------------- node kernel: h = relu(aggP*dinv + b1 + R); q = h.W2l; r2 = h.W2r ----------------
__global__ void __launch_bounds__(256)
node1_kernel(const float* __restrict__ aggP, const float* __restrict__ R,
             const float* __restrict__ deg, const float* __restrict__ b1,
             const float* __restrict__ W2l, const float* __restrict__ W2r,
             float* __restrict__ q, float* __restrict__ r2, int n) {
    const int lane = threadIdx.x & 31;
    int i = blockIdx.x * 8 + (threadIdx.x >> 5);
    if (i >= n) return;
    float dg   = deg[i];
    float dinv = dg > 0.0f ? 1.0f / dg : 0.0f;
    int c = lane * 2;
    float2 ag = *(const float2*)&aggP[(long)i * 64 + c];
    float2 rr = *(const float2*)&R[(long)i * 64 + c];
    float2 bb = *(const float2*)&b1[c];
    float h0 = ag.x * dinv + bb.x + rr.x; h0 = h0 > 0.0f ? h0 : 0.0f;
    float h1 = ag.y * dinv + bb.y + rr.y; h1 = h1 > 0.0f ? h1 : 0.0f;
    float2 wl = *(const float2*)&W2l[c];
    float2 wr = *(const float2*)&W2r[c];
    float ql = h0 * wl.x + h1 * wl.y;
    float qr = h0 * wr.x + h1 * wr.y;
#pragma unroll
    for (int off = 16; off > 0; off >>= 1) {   // wave32 butterfly reduction
        ql += __shfl_xor(ql, off);
        qr += __shfl_xor(qr, off);
    }
    if (lane == 0) { q[i] = ql; r2[i] = qr; }
}

// ---------------- layer-2 scalar aggregation: qsum[dst] += q[src] ----------------
__global__ void scatter2_kernel(const long long* __restrict__ ei,
                                const float* __restrict__ q,
                                float* __restrict__ qsum, long E) {
    long e = (long)blockIdx.x * blockDim.x + threadIdx.x;
    if (e >= E) return;
    int s = (int)ei[e];
    int d = (int)ei[E + e];
    unsafeAtomicAdd(&qsum[d], q[s]);
}

// ---------------- finalize: sigmoid(dinv*qsum + b2 + r2) ----------------
__global__ void final_kernel(const float* __restrict__ qsum, const float* __restrict__ r2,
                             const float* __restrict__ deg, const float* __restrict__ b2,
                             float* __restrict__ out, int n) {
    int i = blockIdx.x * blockDim.x + threadIdx.x;
    if (i >= n) return;
    float dg   = deg[i];
    float dinv = dg > 0.0f ? 1.0f / dg : 0.0f;
    float z = dinv * qsum[i] + b2[0] + r2[i];
    out[i] = 1.0f / (1.0f + __expf(-z));
}

extern "C" void kernel_launch(void* const* d_in, const int* in_sizes, int n_in,
                              void* d_out, int out_size, void* d_ws, size_t ws_size,
                              hipStream_t stream) {
    const float*     x   = (const float*)d_in[0];
    const long long* ei  = (const long long*)d_in[1];
    const float*     W1l = (const float*)d_in[2];
    const float*     b1  = (const float*)d_in[3];
    const float*     W1r = (const float*)d_in[4];
    const float*     W2l = (const float*)d_in[5];
    const float*     b2  = (const float*)d_in[6];
    const float*     W2r = (const float*)d_in[7];
    float*           out = (float*)d_out;

    const int  n = in_sizes[0] / 128;
    const long E = (long)in_sizes[1] / 2;

    // workspace layout (floats): [deg n][aggP 64n][qsum n] | [P 64n][R 64n][q n][r2 n]
    float* W    = (float*)d_ws;
    float* deg  = W;
    float* aggP = W + (long)n;
    float* qsum = W + (long)65 * n;
    float* P    = W + (long)66 * n;
    float* R    = W + (long)130 * n;
    float* q    = W + (long)194 * n;
    float* r2   = W + (long)195 * n;

    zero_kernel<<<2048, 256, 0, stream>>>(W, (long)66 * n);               // deg+aggP+qsum
    gemm1_kernel<<<(n + 63) / 64, 128, 0, stream>>>(x, W1l, W1r, P, R, n);
    scatter1_kernel<<<3200, 256, 0, stream>>>(ei, P, aggP, deg, E);
    node1_kernel<<<(n + 7) / 8, 256, 0, stream>>>(aggP, R, deg, b1, W2l, W2r, q, r2, n);
    scatter2_kernel<<<(int)((E + 255) / 256), 256, 0, stream>>>(ei, q, qsum, E);
    final_kernel<<<(n + 255) / 256, 256, 0, stream>>>(qsum, r2, deg, b2, out, n);
}